// GAT_13091060319067
// MI455X (gfx1250) — compile-verified
//
#include <hip/hip_runtime.h>
#include <stdint.h>

typedef __attribute__((ext_vector_type(2))) float v2f;
typedef __attribute__((ext_vector_type(8))) float v8f;

#define NEG_SLOPE 0.2f
#define BN_EPS 1e-5f

#if defined(__has_builtin) && __has_builtin(__builtin_amdgcn_sched_barrier)
#define SCHED_FENCE() __builtin_amdgcn_sched_barrier(0)
#else
#define SCHED_FENCE()
#endif

// ---------- order-preserving float<->uint encoding for atomicMax ----------
__device__ __forceinline__ unsigned fenc(float f) {
    unsigned u = __float_as_uint(f);
    return (u & 0x80000000u) ? ~u : (u | 0x80000000u);
}
__device__ __forceinline__ float fdec(unsigned u) {
    return (u & 0x80000000u) ? __uint_as_float(u ^ 0x80000000u)
                             : __uint_as_float(~u);
}

// ---------------- WMMA fp32 GEMM: Y[n,M] = X[n,K] @ W[M,K]^T (+bias) -------
// One wave computes a 16 x (16*MT) output tile via V_WMMA_F32_16X16X4_F32.
// Ping-pong pipelined: fragment set for step s+1 is loaded (behind a
// sched_barrier fence) before the WMMAs of step s execute, with alternating
// register sets and no copies, so VMEM latency overlaps the matrix pipe.
// GUARD=true handles M not a multiple of 16 (layer 2, M=47) branchlessly so
// EXEC stays all-ones at every WMMA.
template <int MT, bool GUARD>
__device__ __forceinline__ void load_frag(const float* __restrict__ xr,
                                          const float* const (&wr)[MT],
                                          const bool (&cok)[MT],
                                          int k, v2f& a, v2f (&b)[MT]) {
    a.x = xr[k];
    a.y = xr[k + 1];
    #pragma unroll
    for (int t = 0; t < MT; ++t) {
        float b0 = wr[t][k], b1 = wr[t][k + 1];
        b[t].x = GUARD ? (cok[t] ? b0 : 0.0f) : b0;
        b[t].y = GUARD ? (cok[t] ? b1 : 0.0f) : b1;
    }
}

template <int MT>
__device__ __forceinline__ void wmma_group(const v2f& a, const v2f (&b)[MT],
                                           v8f (&acc)[MT]) {
    #pragma unroll
    for (int t = 0; t < MT; ++t)
        acc[t] = __builtin_amdgcn_wmma_f32_16x16x4_f32(
                     false, a, false, b[t], (short)0, acc[t], false, false);
}

template <int MT, bool GUARD>
__global__ void wmma_gemm_kernel(const float* __restrict__ X,
                                 const float* __restrict__ W,
                                 const float* __restrict__ bias,
                                 float* __restrict__ Y,
                                 int n, int K, int M, int mGroups) {
    int gid  = blockIdx.x * blockDim.x + threadIdx.x;
    int wave = gid >> 5;
    int lane = gid & 31;
    int nTiles = n >> 4;                     // n is a multiple of 16 (40000)
    if (wave >= nTiles * mGroups) return;    // wave-uniform exit
    int tn = wave / mGroups;
    int tg = wave % mGroups;

    int arow = (tn << 4) + (lane & 15);      // node row fed into A
    int kb   = (lane >> 4) << 1;             // lanes 0-15: K 0,1  lanes 16-31: K 2,3
    const float* xr = X + (size_t)arow * K + kb;

    int bcol[MT];
    bool cok[MT];
    const float* wr[MT];
    #pragma unroll
    for (int t = 0; t < MT; ++t) {
        bcol[t] = ((tg * MT + t) << 4) + (lane & 15);
        cok[t]  = GUARD ? (bcol[t] < M) : true;
        wr[t]   = W + (size_t)(cok[t] ? bcol[t] : 0) * K + kb;
    }

    v2f a0, a1;
    v2f b0[MT], b1[MT];
    v8f acc[MT] = {};

    load_frag<MT, GUARD>(xr, wr, cok, 0, a0, b0);   // step 0 -> set 0

    int kSteps = K >> 2;   // even for K = 128 / 256
    int s = 0;
    for (; s + 2 < kSteps; s += 2) {
        load_frag<MT, GUARD>(xr, wr, cok, (s + 1) << 2, a1, b1);
        SCHED_FENCE();
        wmma_group<MT>(a0, b0, acc);
        SCHED_FENCE();
        load_frag<MT, GUARD>(xr, wr, cok, (s + 2) << 2, a0, b0);
        SCHED_FENCE();
        wmma_group<MT>(a1, b1, acc);
        SCHED_FENCE();
    }
    if (s + 1 < kSteps) {      // 2-step tail (kSteps even)
        load_frag<MT, GUARD>(xr, wr, cok, (s + 1) << 2, a1, b1);
        wmma_group<MT>(a0, b0, acc);
        wmma_group<MT>(a1, b1, acc);
    } else {                   // 1-step tail (safety)
        wmma_group<MT>(a0, b0, acc);
    }

    int rbase = (tn << 4) + ((lane >> 4) << 3);  // M = v + 8*(lane/16)
    #pragma unroll
    for (int t = 0; t < MT; ++t) {
        if (!GUARD || bcol[t] < M) {
            float bv = bias ? bias[bcol[t]] : 0.0f;
            #pragma unroll
            for (int v = 0; v < 8; ++v)
                Y[(size_t)(rbase + v) * M + bcol[t]] = acc[t][v] + bv;
        }
    }
}

// ---------------- identity residual init: Y = X + bias ----------------
__global__ void add_bias_kernel(const float* __restrict__ X,
                                const float* __restrict__ bias,
                                float* __restrict__ Y, int total, int M) {
    int i = blockIdx.x * blockDim.x + threadIdx.x;
    if (i >= total) return;
    Y[i] = X[i] + bias[i % M];
}

// ---------------- attention scores: el/er [n, heads] ----------------
__global__ void attn_scores_kernel(const float* __restrict__ feat,
                                   const float* __restrict__ al,
                                   const float* __restrict__ ar,
                                   float* __restrict__ el,
                                   float* __restrict__ er,
                                   int n, int heads, int out) {
    int idx = blockIdx.x * blockDim.x + threadIdx.x;
    if (idx >= n * heads) return;
    int node = idx / heads, h = idx % heads;
    const float* f  = feat + (size_t)node * heads * out + h * out;
    const float* wl = al + h * out;
    const float* wr = ar + h * out;
    float sl = 0.f, sr = 0.f;
    for (int o = 0; o < out; ++o) {
        float v = f[o];
        sl += v * wl[o];
        sr += v * wr[o];
    }
    el[idx] = sl;
    er[idx] = sr;
}

// ---------------- edge pass 1: e = leaky(el[src]+er[dst]); segment max ------
__global__ void edge_max_kernel(const int* __restrict__ src,
                                const int* __restrict__ dst,
                                const float* __restrict__ el,
                                const float* __restrict__ er,
                                float* __restrict__ ebuf,
                                unsigned* __restrict__ m,
                                int E, int heads) {
    int idx = blockIdx.x * blockDim.x + threadIdx.x;
    if (idx >= E * heads) return;
    int e = idx / heads, h = idx % heads;
    int s = src[e], d = dst[e];
    float v = el[s * heads + h] + er[d * heads + h];
    v = (v > 0.f) ? v : NEG_SLOPE * v;
    ebuf[idx] = v;
    atomicMax(&m[d * heads + h], fenc(v));
}

// ---------------- edge pass 2: a = exp(e - m[dst]); segment sum ----------
__global__ void edge_exp_kernel(const int* __restrict__ dst,
                                float* __restrict__ ebuf,
                                const unsigned* __restrict__ m,
                                float* __restrict__ denom,
                                int E, int heads) {
    int idx = blockIdx.x * blockDim.x + threadIdx.x;
    if (idx >= E * heads) return;
    int e = idx / heads, h = idx % heads;
    int d = dst[e];
    float a = __expf(ebuf[idx] - fdec(m[d * heads + h]));
    ebuf[idx] = a;
    atomicAdd(&denom[d * heads + h], a);
}

// ---------------- edge pass 3: rst[dst] += feat[src] * alpha ----------
__global__ void edge_aggregate_kernel(const int* __restrict__ src,
                                      const int* __restrict__ dst,
                                      const float* __restrict__ ebuf,
                                      const float* __restrict__ denom,
                                      const float* __restrict__ feat,
                                      float* __restrict__ rst,
                                      int E, int heads, int out) {
    int idx = blockIdx.x * blockDim.x + threadIdx.x;
    if (idx >= E * heads) return;
    int e = idx / heads, h = idx % heads;
    int s = src[e], d = dst[e];
    float alpha = ebuf[idx] / denom[d * heads + h];
    const float* fs = feat + (size_t)s * heads * out + h * out;
    float*       rd = rst  + (size_t)d * heads * out + h * out;
    for (int o = 0; o < out; ++o)
        atomicAdd(&rd[o], fs[o] * alpha);
}

// ---------------- batch norm: per-channel mean/var (block per channel) ----
__global__ void bn_stats_kernel(const float* __restrict__ x,
                                float* __restrict__ stats, int n, int M) {
    int c = blockIdx.x;
    float s = 0.f, sq = 0.f;
    for (int i = threadIdx.x; i < n; i += blockDim.x) {
        float v = x[(size_t)i * M + c];
        s += v;
        sq += v * v;
    }
    __shared__ float ss[256], ssq[256];
    ss[threadIdx.x] = s;
    ssq[threadIdx.x] = sq;
    __syncthreads();
    for (int st = blockDim.x >> 1; st > 0; st >>= 1) {
        if (threadIdx.x < (unsigned)st) {
            ss[threadIdx.x] += ss[threadIdx.x + st];
            ssq[threadIdx.x] += ssq[threadIdx.x + st];
        }
        __syncthreads();
    }
    if (threadIdx.x == 0) {
        float mu = ss[0] / (float)n;
        float var = ssq[0] / (float)n - mu * mu;
        stats[2 * c]     = mu;
        stats[2 * c + 1] = var;
    }
}

__global__ void bn_apply_relu_kernel(float* __restrict__ x,
                                     const float* __restrict__ stats,
                                     const float* __restrict__ gamma,
                                     const float* __restrict__ beta,
                                     int total, int M) {
    int i = blockIdx.x * blockDim.x + threadIdx.x;
    if (i >= total) return;
    int c = i % M;
    float mu = stats[2 * c], var = stats[2 * c + 1];
    float v = gamma[c] * (x[i] - mu) * __frsqrt_rn(var + BN_EPS) + beta[c];
    x[i] = (v > 0.f) ? v : 0.f;
}

// =========================================================================
extern "C" void kernel_launch(void* const* d_in, const int* in_sizes, int n_in,
                              void* d_out, int out_size, void* d_ws, size_t ws_size,
                              hipStream_t stream) {
    const int IN = 128, H = 64, HEADS = 4, C = 47, HH = HEADS * H;  // 256
    const int N = in_sizes[0] / IN;
    const int E = in_sizes[1];

    const float* feat    = (const float*)d_in[0];
    const int*   src     = (const int*)d_in[1];
    const int*   dst     = (const int*)d_in[2];
    const float* fc_w0   = (const float*)d_in[3];
    const float* attn_l0 = (const float*)d_in[4];
    const float* attn_r0 = (const float*)d_in[5];
    const float* bias0   = (const float*)d_in[6];
    const float* res_w0  = (const float*)d_in[7];
    const float* gamma0  = (const float*)d_in[8];
    const float* beta0   = (const float*)d_in[9];
    const float* fc_w1   = (const float*)d_in[10];
    const float* attn_l1 = (const float*)d_in[11];
    const float* attn_r1 = (const float*)d_in[12];
    const float* bias1   = (const float*)d_in[13];
    const float* gamma1  = (const float*)d_in[14];
    const float* beta1   = (const float*)d_in[15];
    const float* fc_w2   = (const float*)d_in[16];
    const float* attn_l2 = (const float*)d_in[17];
    const float* attn_r2 = (const float*)d_in[18];
    const float* bias2   = (const float*)d_in[19];
    const float* res_w2  = (const float*)d_in[20];
    float* out = (float*)d_out;

    // -------- workspace layout (floats) --------
    float* ws = (float*)d_ws;
    size_t off = 0;
    float*    fcbuf = ws + off; off += (size_t)N * HH;   // fc output (feat)
    float*    bufA  = ws + off; off += (size_t)N * HH;   // layer0 output
    float*    bufB  = ws + off; off += (size_t)N * HH;   // layer1 output
    float*    el    = ws + off; off += (size_t)N * HEADS;
    float*    er    = ws + off; off += (size_t)N * HEADS;
    unsigned* mbuf  = (unsigned*)(ws + off); off += (size_t)N * HEADS;
    float*    denom = ws + off; off += (size_t)N * HEADS;
    float*    ebuf  = ws + off; off += (size_t)E * HEADS;
    float*    bnst  = ws + off; off += 2 * HH;

    const int TB = 256;
    auto blocks = [](long long t, int b) { return (int)((t + b - 1) / b); };

    // GEMM dispatch: M multiple of 64 -> fat 16x64 tiles, no guard;
    // otherwise 16x16 tiles with branchless column guard (M=47).
    auto gemm = [&](const float* X, const float* W, const float* bias, float* Y,
                    int n, int K, int M) {
        if ((M & 63) == 0) {
            int mGroups = M / 64;
            long long waves = (long long)(n / 16) * mGroups;
            wmma_gemm_kernel<4, false><<<blocks(waves * 32, TB), TB, 0, stream>>>(
                X, W, bias, Y, n, K, M, mGroups);
        } else {
            int mGroups = (M + 15) / 16;
            long long waves = (long long)(n / 16) * mGroups;
            wmma_gemm_kernel<1, true><<<blocks(waves * 32, TB), TB, 0, stream>>>(
                X, W, bias, Y, n, K, M, mGroups);
        }
    };
    auto gat_edges = [&](const float* featp, const float* al, const float* ar,
                         float* rst, int heads, int outd) {
        long long nh = (long long)N * heads;
        long long eh = (long long)E * heads;
        hipMemsetAsync(mbuf, 0, nh * sizeof(unsigned), stream);
        hipMemsetAsync(denom, 0, nh * sizeof(float), stream);
        attn_scores_kernel<<<blocks(nh, TB), TB, 0, stream>>>(
            featp, al, ar, el, er, N, heads, outd);
        edge_max_kernel<<<blocks(eh, TB), TB, 0, stream>>>(
            src, dst, el, er, ebuf, mbuf, E, heads);
        edge_exp_kernel<<<blocks(eh, TB), TB, 0, stream>>>(
            dst, ebuf, mbuf, denom, E, heads);
        edge_aggregate_kernel<<<blocks(eh, TB), TB, 0, stream>>>(
            src, dst, ebuf, denom, featp, rst, E, heads, outd);
    };
    auto bn_relu = [&](float* x, const float* g, const float* b, int M) {
        bn_stats_kernel<<<M, 256, 0, stream>>>(x, bnst, N, M);
        bn_apply_relu_kernel<<<blocks((long long)N * M, TB), TB, 0, stream>>>(
            x, bnst, g, b, N * M, M);
    };

    // ---------------- layer 0: in=128, heads=4, out=64, linear residual ----
    gemm(feat, fc_w0, nullptr, fcbuf, N, IN, HH);        // feat transform
    gemm(feat, res_w0, bias0, bufA, N, IN, HH);          // accumulator = res + bias
    gat_edges(fcbuf, attn_l0, attn_r0, bufA, HEADS, H);  // += sum(alpha * feat[src])
    bn_relu(bufA, gamma0, beta0, HH);

    // ---------------- layer 1: in=256, heads=4, out=64, identity residual --
    gemm(bufA, fc_w1, nullptr, fcbuf, N, HH, HH);
    add_bias_kernel<<<blocks((long long)N * HH, TB), TB, 0, stream>>>(
        bufA, bias1, bufB, N * HH, HH);                  // accumulator = h + bias
    gat_edges(fcbuf, attn_l1, attn_r1, bufB, HEADS, H);
    bn_relu(bufB, gamma1, beta1, HH);

    // ---------------- layer 2: in=256, heads=1, out=47, linear residual ----
    gemm(bufB, fc_w2, nullptr, fcbuf, N, HH, C);         // feat2 [N,47]
    gemm(bufB, res_w2, bias2, out, N, HH, C);            // d_out = res + bias
    gat_edges(fcbuf, attn_l2, attn_r2, out, 1, C);       // mean over 1 head == identity
}